// RelGraphConvLayer_1760936591781
// MI455X (gfx1250) — compile-verified
//
#include <hip/hip_runtime.h>

// Problem constants (match reference)
#define NN     100000
#define RR     8
#define EE     150000
#define DD     128
#define LDS_STRIDE 388   // 384 + 4 pad: 384 % 64 == 0 would put all 16 rows in one bank

typedef __attribute__((ext_vector_type(2))) float v2f;
typedef __attribute__((ext_vector_type(8))) float v8f;

// ---------------- zero workspace ----------------
__global__ void zero_kernel(float* __restrict__ p, long n) {
    long i = (long)blockIdx.x * blockDim.x + threadIdx.x;
    long stride = (long)gridDim.x * blockDim.x;
    for (; i < n; i += stride) p[i] = 0.0f;
}

// ---------------- per-(relation,node) in-degree ----------------
__global__ void degree_kernel(const int* __restrict__ dst, float* __restrict__ deg) {
    int i = blockIdx.x * blockDim.x + threadIdx.x;
    if (i >= RR * EE) return;
    int r = i / EE;
    atomicAdd(&deg[(long)r * NN + dst[i]], 1.0f);
}

// ---------------- basis-folded normalized scatter ----------------
// 32 lanes per edge, 4 dims per lane. Accumulates c[r][b]/deg * x[src]
// into aggB[b] (b=0,1). aggB (102MB) + x (51MB) stay L2-resident.
__global__ void scatter_kernel(const float* __restrict__ x,
                               const int* __restrict__ src,
                               const int* __restrict__ dst,
                               const float* __restrict__ coeff,
                               const float* __restrict__ deg,
                               float* __restrict__ agg0,
                               float* __restrict__ agg1) {
    long t  = (long)blockIdx.x * blockDim.x + threadIdx.x;
    long eg = t >> 5;
    if (eg >= (long)RR * EE) return;
    int lane = (int)(t & 31);
    int r = (int)(eg / EE);
    int s = src[eg];
    int d = dst[eg];
    float inv = 1.0f / fmaxf(deg[(long)r * NN + d], 1.0f);
    float c0 = coeff[r * 2 + 0] * inv;
    float c1 = coeff[r * 2 + 1] * inv;
    const float4 xv = *(const float4*)(x + (long)s * DD + lane * 4);
    float* p0 = agg0 + (long)d * DD + lane * 4;
    float* p1 = agg1 + (long)d * DD + lane * 4;
    atomicAdd(p0 + 0, xv.x * c0); atomicAdd(p0 + 1, xv.y * c0);
    atomicAdd(p0 + 2, xv.z * c0); atomicAdd(p0 + 3, xv.w * c0);
    atomicAdd(p1 + 0, xv.x * c1); atomicAdd(p1 + 1, xv.y * c1);
    atomicAdd(p1 + 2, xv.z * c1); atomicAdd(p1 + 3, xv.w * c1);
}

// ---------------- fused WMMA GEMM + bias + ReLU ----------------
// h[n,:] = aggB0[n]·bases[0] + aggB1[n]·bases[1] + x[n]·W_loop  (K = 384)
// One block = 16 nodes x 128 cols; 8 waves, each wave one 16x16 tile via
// 96 x V_WMMA_F32_16X16X4_F32. A panel staged in LDS.
__global__ __launch_bounds__(256) void gemm_kernel(
        const float* __restrict__ agg0, const float* __restrict__ agg1,
        const float* __restrict__ x,    const float* __restrict__ bases,
        const float* __restrict__ loopw, const float* __restrict__ bias,
        float* __restrict__ out) {
    __shared__ float As[16 * LDS_STRIDE];
    const int t = threadIdx.x;
    const int node0 = blockIdx.x * 16;

    // Stage A: 16 rows x 384 (k<128: aggB0, k<256: aggB1, else x). Coalesced float4.
    const float* segs[3] = {agg0, agg1, x};
    for (int seg = 0; seg < 3; ++seg) {
        const float* sp = segs[seg];
        for (int i = t; i < 16 * 32; i += 256) {
            int row = i >> 5;
            int c4  = (i & 31) * 4;
            float4 v = *(const float4*)(sp + (long)(node0 + row) * DD + c4);
            float* dp = &As[row * LDS_STRIDE + seg * 128 + c4];
            dp[0] = v.x; dp[1] = v.y; dp[2] = v.z; dp[3] = v.w;
        }
    }
    __syncthreads();

    const int wave = t >> 5;
    const int lane = t & 31;
    const int half = lane >> 4;      // K-half selector (ISA 16x4 A / 4x16 B layout)
    const int l16  = lane & 15;      // A: row m; B: col n
    const int n    = wave * 16 + l16;

    v8f acc = {};
    const float* W[3] = {bases, bases + 128 * 128, loopw};
    #pragma unroll
    for (int seg = 0; seg < 3; ++seg) {
        const float* Wp = W[seg];
        #pragma unroll 8
        for (int kk = 0; kk < 32; ++kk) {
            const int k0   = kk * 4;                 // k within segment
            const int krow = k0 + 2 * half;
            v2f a, b;
            const float* ap = &As[l16 * LDS_STRIDE + seg * 128 + k0 + 2 * half];
            a.x = ap[0];            a.y = ap[1];              // A[m][krow], A[m][krow+1]
            b.x = Wp[krow * DD + n]; b.y = Wp[(krow + 1) * DD + n];
            acc = __builtin_amdgcn_wmma_f32_16x16x4_f32(
                false, a, false, b, (short)0, acc, false, false);
        }
    }

    // C layout: VGPR j -> M = j + 8*half, N = l16
    const float bv = bias[n];
    #pragma unroll
    for (int j = 0; j < 8; ++j) {
        int node = node0 + j + 8 * half;
        float v = acc[j] + bv;
        out[(long)node * DD + n] = v > 0.0f ? v : 0.0f;
    }
}

extern "C" void kernel_launch(void* const* d_in, const int* in_sizes, int n_in,
                              void* d_out, int out_size, void* d_ws, size_t ws_size,
                              hipStream_t stream) {
    const float* x     = (const float*)d_in[0];
    const int*   src   = (const int*)  d_in[1];
    const int*   dst   = (const int*)  d_in[2];
    const float* coeff = (const float*)d_in[3];
    const float* bases = (const float*)d_in[4];
    const float* loopw = (const float*)d_in[5];
    const float* bias  = (const float*)d_in[6];
    float* out = (float*)d_out;

    // Workspace layout: aggB0 | aggB1 | deg  (~105.6 MB total)
    float* agg0 = (float*)d_ws;
    float* agg1 = agg0 + (long)NN * DD;
    float* deg  = agg1 + (long)NN * DD;
    long zcount = 2L * NN * DD + (long)RR * NN;

    zero_kernel<<<4096, 256, 0, stream>>>(agg0, zcount);

    degree_kernel<<<(RR * EE + 255) / 256, 256, 0, stream>>>(dst, deg);

    long sthreads = (long)RR * EE * 32;
    scatter_kernel<<<(int)((sthreads + 255) / 256), 256, 0, stream>>>(
        x, src, dst, coeff, deg, agg0, agg1);

    gemm_kernel<<<NN / 16, 256, 0, stream>>>(agg0, agg1, x, bases, loopw, bias, out);
}